// VertexNormalsPYG_57037165691509
// MI455X (gfx1250) — compile-verified
//
#include <hip/hip_runtime.h>

#define EPS_N 1e-12f

// ---------------------------------------------------------------------------
// Kernel 1: zero the accumulation buffer with 128-bit stores.
// n is the total float count; n4 = n/4 full float4 chunks + scalar tail.
// ---------------------------------------------------------------------------
__global__ void vn_zero_kernel(float4* __restrict__ out4,
                               float*  __restrict__ out,
                               int n4, int n) {
    int i = blockIdx.x * blockDim.x + threadIdx.x;
    if (i < n4) {
        out4[i] = make_float4(0.0f, 0.0f, 0.0f, 0.0f);   // global_store_b128
    }
    // scalar tail (at most 3 elements), handled by the first few threads
    int t = 4 * n4 + i;
    if (i < 4 && t < n) out[t] = 0.0f;
}

// ---------------------------------------------------------------------------
// Kernel 2: per-face normal computation + scatter-add into vertex normals.
// faces: [n_faces,3] int32, v: [n_verts,3] f32, vn: [n_verts,3] f32 accum.
// The 12MB vertex array is fully L2-resident (192MB L2), so the random
// gathers and the 9 global_atomic_add_f32 per face run at L2 speed.
// ---------------------------------------------------------------------------
__global__ void vn_face_scatter_kernel(const float* __restrict__ v,
                                       const int*   __restrict__ faces,
                                       float* __restrict__ vn,
                                       int n_faces) {
    int f = blockIdx.x * blockDim.x + threadIdx.x;
    if (f >= n_faces) return;

    // Prefetch the index stream ahead (gfx1250 global_prefetch_b8,
    // speculative: OOB prefetches are silently dropped).
    __builtin_prefetch(&faces[3 * (f + 2048)], 0, 0);

    // 12-byte index load -> global_load_b96
    int i0 = faces[3 * f + 0];
    int i1 = faces[3 * f + 1];
    int i2 = faces[3 * f + 2];

    // Three 12-byte vertex gathers -> global_load_b96 (L2 hits)
    float ax = v[3 * i0 + 0], ay = v[3 * i0 + 1], az = v[3 * i0 + 2];
    float bx = v[3 * i1 + 0], by = v[3 * i1 + 1], bz = v[3 * i1 + 2];
    float cx = v[3 * i2 + 0], cy = v[3 * i2 + 1], cz = v[3 * i2 + 2];

    // Edges, exactly as reference: e0 = v1-v0, e1 = v2-v1, e2 = v0-v2
    float e0x = bx - ax, e0y = by - ay, e0z = bz - az;
    float e1x = cx - bx, e1y = cy - by, e1z = cz - bz;
    float e2x = ax - cx, e2y = ay - cy, e2z = az - cz;

    // cross(e0,e1) + cross(e1,e2) + cross(e2,e0): algebraically 3*cross(e0,e1)
    // but NOT bitwise-equal in fp32, so compute all three like the reference.
    float c0x = e0y * e1z - e0z * e1y;
    float c0y = e0z * e1x - e0x * e1z;
    float c0z = e0x * e1y - e0y * e1x;

    float c1x = e1y * e2z - e1z * e2y;
    float c1y = e1z * e2x - e1x * e2z;
    float c1z = e1x * e2y - e1y * e2x;

    float c2x = e2y * e0z - e2z * e0y;
    float c2y = e2z * e0x - e2x * e0z;
    float c2z = e2x * e0y - e2y * e0x;

    float nx = c0x + c1x + c2x;
    float ny = c0y + c1y + c2y;
    float nz = c0z + c1z + c2z;

    // 9 no-return fp32 atomics -> global_atomic_add_f32 (STOREcnt path, L2 RMW)
    atomicAdd(&vn[3 * i0 + 0], nx);
    atomicAdd(&vn[3 * i0 + 1], ny);
    atomicAdd(&vn[3 * i0 + 2], nz);
    atomicAdd(&vn[3 * i1 + 0], nx);
    atomicAdd(&vn[3 * i1 + 1], ny);
    atomicAdd(&vn[3 * i1 + 2], nz);
    atomicAdd(&vn[3 * i2 + 0], nx);
    atomicAdd(&vn[3 * i2 + 1], ny);
    atomicAdd(&vn[3 * i2 + 2], nz);
}

// ---------------------------------------------------------------------------
// Kernel 3: in-place normalization, vn /= max(||vn||, 1e-12).
// Each thread handles 4 vertices = 48 bytes = 3 x float4, so all memory
// traffic is aligned 128-bit loads/stores (global_load_b128/store_b128).
// ---------------------------------------------------------------------------
__device__ __forceinline__ void vn_norm3(float& x, float& y, float& z) {
    float nrm = sqrtf(x * x + y * y + z * z);
    float inv = 1.0f / fmaxf(nrm, EPS_N);
    x *= inv; y *= inv; z *= inv;
}

__global__ void vn_normalize_kernel(float* __restrict__ vn, int n_verts) {
    int i = blockIdx.x * blockDim.x + threadIdx.x;   // group of 4 vertices
    int v0 = 4 * i;
    if (v0 >= n_verts) return;

    if (v0 + 3 < n_verts) {
        float4* p = (float4*)(vn + 3 * v0);          // 48B chunk, 16B aligned
        float4 a = p[0];                              // v0.xyz, v1.x
        float4 b = p[1];                              // v1.yz,  v2.xy
        float4 c = p[2];                              // v2.z,   v3.xyz

        vn_norm3(a.x, a.y, a.z);                      // vertex 0
        vn_norm3(a.w, b.x, b.y);                      // vertex 1
        vn_norm3(b.z, b.w, c.x);                      // vertex 2
        vn_norm3(c.y, c.z, c.w);                      // vertex 3

        p[0] = a; p[1] = b; p[2] = c;
    } else {
        // ragged tail: scalar per-vertex path
        for (int k = v0; k < n_verts; ++k) {
            float x = vn[3 * k + 0];
            float y = vn[3 * k + 1];
            float z = vn[3 * k + 2];
            vn_norm3(x, y, z);
            vn[3 * k + 0] = x;
            vn[3 * k + 1] = y;
            vn[3 * k + 2] = z;
        }
    }
}

extern "C" void kernel_launch(void* const* d_in, const int* in_sizes, int n_in,
                              void* d_out, int out_size, void* d_ws, size_t ws_size,
                              hipStream_t stream) {
    const float* v     = (const float*)d_in[0];   // [n_verts, 3] f32
    const int*   faces = (const int*)d_in[1];     // [n_faces, 3] i32
    float*       vn    = (float*)d_out;           // [n_verts, 3] f32

    const int n_verts = in_sizes[0] / 3;
    const int n_faces = in_sizes[1] / 3;
    const int n_out   = out_size;                 // n_verts * 3
    const int n_out4  = n_out / 4;                // full float4 chunks

    const int BLK = 256;  // 8 wave32 waves per block

    vn_zero_kernel<<<(n_out4 + BLK - 1) / BLK, BLK, 0, stream>>>(
        (float4*)vn, vn, n_out4, n_out);

    vn_face_scatter_kernel<<<(n_faces + BLK - 1) / BLK, BLK, 0, stream>>>(
        v, faces, vn, n_faces);

    const int n_grp = (n_verts + 3) / 4;          // 4 vertices per thread
    vn_normalize_kernel<<<(n_grp + BLK - 1) / BLK, BLK, 0, stream>>>(
        vn, n_verts);
}